// EncoderWithoutFlexFrontsGraphsage_26139170964034
// MI455X (gfx1250) — compile-verified
//
#include <hip/hip_runtime.h>
#include <hip/hip_bf16.h>
#include <stddef.h>

// GraphSAGE 3-layer forward for MI455X (gfx1250, wave32).
// GEMMs use V_WMMA_F32_16X16X4_F32 (full fp32, matches reference precision).
// Edge scatter uses global_atomic_add_f32 into an L2-resident agg buffer.

typedef float v2f __attribute__((ext_vector_type(2)));
typedef float v8f __attribute__((ext_vector_type(8)));

#define D 64
// LDS row stride (in floats) for one K-pair row: 128 data + 32 pad so that the
// second half-wave (p+1) lands on banks 32..63 -> fully conflict-free b64 loads.
#define WSTRIDE 160

// ---------------------------------------------------------------- utilities
__global__ void zero_f32_kernel(float* __restrict__ p, int n) {
    int i = blockIdx.x * blockDim.x + threadIdx.x;
    if (i < n) p[i] = 0.0f;
}

// one thread per edge: deg[dst] += 1
__global__ void degree_kernel(const int* __restrict__ dst, float* __restrict__ deg, int E) {
    int e = blockIdx.x * blockDim.x + threadIdx.x;
    if (e < E) atomicAdd(deg + dst[e], 1.0f);
}

// in place: deg -> 1/max(deg,1)
__global__ void invdeg_kernel(float* __restrict__ deg, int n) {
    int i = blockIdx.x * blockDim.x + threadIdx.x;
    if (i < n) deg[i] = 1.0f / fmaxf(deg[i], 1.0f);
}

// one thread per (edge, feature): agg[dst][f] += x[src][f]
// block = 256 threads = 4 edges x 64 features, fully coalesced per edge row.
__global__ void scatter_add_kernel(const float* __restrict__ x,
                                   const int* __restrict__ src,
                                   const int* __restrict__ dst,
                                   float* __restrict__ agg, int E) {
    int tid = blockIdx.x * 256 + threadIdx.x;
    int e = tid >> 6;
    int f = tid & 63;
    if (e < E) {
        int s = src[e];
        int d = dst[e];
        atomicAdd(agg + (size_t)d * D + f, x[(size_t)s * D + f]);
    }
}

// ------------------------------------------------- fused SAGEConv layer GEMM
// out[n,:] = relu( (agg[n,:]*invdeg[n]) @ Wl^T + x[n,:] @ Wr^T + bl )
// 256 threads = 8 waves; each wave computes a 16-row x 64-col tile via
// v_wmma_f32_16x16x4_f32, K-loop over 64 in steps of 4.
__global__ void __launch_bounds__(256)
sage_gemm_relu_kernel(const float* __restrict__ xin,
                      const float* __restrict__ agg,
                      const float* __restrict__ invdeg,
                      const float* __restrict__ Wl,   // [64][64] row-major (j,k)
                      const float* __restrict__ bl,   // [64]
                      const float* __restrict__ Wr,   // [64][64]
                      float* __restrict__ xout, int N) {
    // Stage weights K-pair interleaved:
    //   sW[p*WSTRIDE + j*2 + q] = W[j*64 + 2p + q],  p = k>>1, q = k&1
    // so each lane's B operand {B[k][j], B[k+1][j]} is ONE aligned ds_load_b64.
    __shared__ float sWl[32 * WSTRIDE];
    __shared__ float sWr[32 * WSTRIDE];
    for (int idx = threadIdx.x; idx < D * D; idx += 256) {
        int p = idx >> 7;        // K pair (0..31)
        int r = idx & 127;       // j*2 + q
        int j = r >> 1;
        int q = r & 1;
        sWl[p * WSTRIDE + r] = Wl[j * D + 2 * p + q];
        sWr[p * WSTRIDE + r] = Wr[j * D + 2 * p + q];
    }
    __syncthreads();

    const int lane = threadIdx.x & 31;
    const int wave = threadIdx.x >> 5;
    const int mrow = lane & 15;        // A: M row / B: N col / C: N col
    const int half = lane >> 4;        // selects K pair (A,B) or M+8 (C/D)

    const int rowbase = blockIdx.x * 128 + wave * 16;
    // Clamp row for loads so EXEC stays all-1s through WMMA (ISA requirement);
    // out-of-range rows are discarded at the guarded store.
    int arow = rowbase + mrow;
    if (arow > N - 1) arow = N - 1;

    const float* __restrict__ aggrow = agg + (size_t)arow * D;
    const float* __restrict__ xrow   = xin + (size_t)arow * D;
    const float invd = invdeg[arow];

    v8f acc[4];
#pragma unroll
    for (int t = 0; t < 4; ++t) acc[t] = (v8f){0.f, 0.f, 0.f, 0.f, 0.f, 0.f, 0.f, 0.f};

#pragma unroll 4
    for (int k0 = 0; k0 < D; k0 += 4) {
        // this lane's K pair index: p = k0/2 + half  ->  k = {2p, 2p+1}
        const int prow = ((k0 >> 1) + half) * WSTRIDE + mrow * 2;
        const int ka = k0 + half * 2;

        // A operands: 16x4 fp32, lanes 0-15 = rows, VGPR pair = K
        v2f aAgg = *(const v2f*)(aggrow + ka);
        aAgg.x *= invd;
        aAgg.y *= invd;
        v2f aX = *(const v2f*)(xrow + ka);

#pragma unroll
        for (int t = 0; t < 4; ++t) {
            // B operands: {B[ka][j], B[ka+1][j]} contiguous -> single b64 load
            v2f bWl = *(const v2f*)(sWl + prow + t * 32);
            v2f bWr = *(const v2f*)(sWr + prow + t * 32);

            acc[t] = __builtin_amdgcn_wmma_f32_16x16x4_f32(
                false, aAgg, false, bWl, (short)0, acc[t], false, false);
            acc[t] = __builtin_amdgcn_wmma_f32_16x16x4_f32(
                false, aX, false, bWr, (short)0, acc[t], false, false);
        }
    }

    // C/D layout: VGPR r -> M = r + half*8 (lanes 0-15: M=r, lanes 16-31: M=r+8),
    // N = lane & 15.
#pragma unroll
    for (int t = 0; t < 4; ++t) {
        const int col = t * 16 + mrow;
        const float bias = bl[col];
#pragma unroll
        for (int r = 0; r < 8; ++r) {
            const int orow = rowbase + r + half * 8;
            if (orow < N) {
                float v = acc[t][r] + bias;
                xout[(size_t)orow * D + col] = fmaxf(v, 0.0f);
            }
        }
    }
}

// ---------------------------------------------------------------- launcher
extern "C" void kernel_launch(void* const* d_in, const int* in_sizes, int n_in,
                              void* d_out, int out_size, void* d_ws, size_t ws_size,
                              hipStream_t stream) {
    const float* x  = (const float*)d_in[0];   // [N, 64]
    const float* Wl = (const float*)d_in[1];   // [3, 64, 64]
    const float* bl = (const float*)d_in[2];   // [3, 64]
    const float* Wr = (const float*)d_in[3];   // [3, 64, 64]
    const int*   ei = (const int*)d_in[4];     // [2, E]

    const int N = in_sizes[0] / D;             // 100000
    const int E = in_sizes[4] / 2;             // 1600000
    const int L = in_sizes[1] / (D * D);       // 3

    const int* src = ei;
    const int* dst = ei + E;

    // Workspace layout (floats): invdeg[N] | agg[N*64] | bufA[N*64] | bufB[N*64]
    float* ws     = (float*)d_ws;
    float* invdeg = ws;
    float* agg    = invdeg + N;
    float* bufA   = agg + (size_t)N * D;
    float* bufB   = bufA + (size_t)N * D;
    float* out    = (float*)d_out;

    const int TB = 256;

    // --- degree -> inv degree (computed once) ---
    zero_f32_kernel<<<(N + TB - 1) / TB, TB, 0, stream>>>(invdeg, N);
    degree_kernel<<<(E + TB - 1) / TB, TB, 0, stream>>>(dst, invdeg, E);
    invdeg_kernel<<<(N + TB - 1) / TB, TB, 0, stream>>>(invdeg, N);

    // --- 3 layers, ping-pong x -> bufA -> bufB -> out ---
    const float* xin[3]  = {x, bufA, bufB};
    float*       xout[3] = {bufA, bufB, out};

    const int aggN       = N * D;
    const int scatterBlk = (E * D + TB - 1) / TB;   // (edge,feature) threads
    const int gemmBlk    = (N + 127) / 128;          // 128 rows per block

    for (int l = 0; l < L; ++l) {
        zero_f32_kernel<<<(aggN + TB - 1) / TB, TB, 0, stream>>>(agg, aggN);
        scatter_add_kernel<<<scatterBlk, TB, 0, stream>>>(xin[l], src, dst, agg, E);
        sage_gemm_relu_kernel<<<gemmBlk, TB, 0, stream>>>(
            xin[l], agg, invdeg,
            Wl + (size_t)l * D * D, bl + (size_t)l * D, Wr + (size_t)l * D * D,
            xout[l], N);
    }
}